// GCN_res_25340307046430
// MI455X (gfx1250) — compile-verified
//
#include <hip/hip_runtime.h>
#include <math.h>

#define F_IN   128
#define HDIM   256
#define NCLS   40
#define NLAY   6
#define BN_EPS 1e-5f

typedef float v2f __attribute__((ext_vector_type(2)));
typedef float v8f __attribute__((ext_vector_type(8)));

// ---------------- degree / normalization ----------------
__global__ void deg_init_kernel(float* __restrict__ deg, int N) {
  int i = blockIdx.x * blockDim.x + threadIdx.x;
  if (i < N) deg[i] = 1.0f;                       // self-loop contributes 1
}

__global__ void deg_scatter_kernel(const int* __restrict__ dstv, float* __restrict__ deg, int E) {
  int e = blockIdx.x * blockDim.x + threadIdx.x;
  if (e < E) atomicAdd(&deg[dstv[e]], 1.0f);
}

__global__ void deg_finalize_kernel(const float* __restrict__ deg,
                                    float* __restrict__ dinv,
                                    float* __restrict__ selfnorm, int N) {
  int i = blockIdx.x * blockDim.x + threadIdx.x;
  if (i < N) {
    float d = deg[i];
    dinv[i]     = rsqrtf(d);
    selfnorm[i] = 1.0f / d;
  }
}

// softmax over the 6 layer weights, done on-device (no host readback allowed)
__global__ void layerw_kernel(const float* __restrict__ lw, float* __restrict__ wvec) {
  if (threadIdx.x == 0 && blockIdx.x == 0) {
    float m = -1e30f;
    for (int l = 0; l < NLAY; ++l) m = fmaxf(m, lw[l]);
    float s = 0.0f;
    float e[NLAY];
    for (int l = 0; l < NLAY; ++l) { e[l] = __expf(lw[l] - m); s += e[l]; }
    float inv = 1.0f / s;
    for (int l = 0; l < NLAY; ++l) wvec[l] = e[l] * inv;
  }
}

__global__ void zero_kernel(float* __restrict__ p, int total) {
  int i = blockIdx.x * blockDim.x + threadIdx.x;
  if (i < total) p[i] = 0.0f;
}

// ---------------- fp32 WMMA GEMM: C[M,256] = A[M,K] @ B[K,256] (+ bias) ----------------
// V_WMMA_F32_16X16X4_F32, K-loop in steps of 4. Block = 256 threads = 8 waves.
// Each wave computes TWO 16x16 output tiles (cols wave*16 and wave*16+128) that SHARE one
// A fragment, so the activation matrix streams from HBM exactly once (grid.y == 1) and
// each A fetch feeds two WMMAs.
// A (16x4 f32, ISA 7.12.2): lanes 0-15 hold M=lane, VGPR0=K0,VGPR1=K1; lanes 16-31 K=2,3.
// B (4x16): VGPR0: lanes0-15 K=0 / lanes16-31 K=2 ; VGPR1: K=1 / K=3 (mirrors C/D striping).
// C/D (16x16 f32): VGPR j: lanes0-15 -> M=j, N=lane ; lanes16-31 -> M=j+8, N=lane-16.
__global__ __launch_bounds__(256)
void gemm_f32_wmma_kernel(const float* __restrict__ A, const float* __restrict__ B,
                          const float* __restrict__ bias, float* __restrict__ C,
                          int M, int K) {
  const int lane    = threadIdx.x & 31;
  const int wave    = threadIdx.x >> 5;
  const int rowbase = blockIdx.x * 16;
  const int col0    = wave * 16 + (lane & 15);
  const int col1    = col0 + 128;
  const int row     = min(rowbase + (lane & 15), M - 1);
  const int klo     = (lane >> 4) << 1;           // 0 for lanes 0-15, 2 for lanes 16-31

  const float* __restrict__ Arow = A + (size_t)row * K;
  v8f acc0 = {}, acc1 = {};

#pragma unroll 4
  for (int k0 = 0; k0 < K; k0 += 4) {
    v2f a, b0, b1;
    a.x = Arow[k0 + klo];
    a.y = Arow[k0 + klo + 1];
    const float* __restrict__ Bk = B + (size_t)(k0 + klo) * HDIM;
    b0.x = Bk[col0];
    b1.x = Bk[col1];
    b0.y = Bk[HDIM + col0];
    b1.y = Bk[HDIM + col1];
    acc0 = __builtin_amdgcn_wmma_f32_16x16x4_f32(false, a, false, b0,
                                                 (short)0, acc0, false, false);
    acc1 = __builtin_amdgcn_wmma_f32_16x16x4_f32(false, a, false, b1,
                                                 (short)0, acc1, false, false);
  }

  const float bv0  = bias ? bias[col0] : 0.0f;
  const float bv1  = bias ? bias[col1] : 0.0f;
  const int   rofs = (lane >> 4) << 3;            // 0 or 8
#pragma unroll
  for (int j = 0; j < 8; ++j) {
    int r = rowbase + rofs + j;
    if (r < M) {
      C[(size_t)r * HDIM + col0] = acc0[j] + bv0;
      C[(size_t)r * HDIM + col1] = acc1[j] + bv1;
    }
  }
}

// ---------------- edge message scatter: agg[dst] += hw[src] * dinv[src]*dinv[dst] -----
// One thread = one edge x 4 channels (float4 gather, 4x global_atomic_add_f32).
// Adjacent lanes cover adjacent float4s of the same edge row -> coalesced gather;
// per-edge scalars are wave-uniform loads.
__global__ __launch_bounds__(256)
void edge_scatter_kernel(const int* __restrict__ srcv, const int* __restrict__ dstv,
                         const float* __restrict__ dinv,
                         const float* __restrict__ hw, float* __restrict__ agg, int E) {
  int idx = blockIdx.x * blockDim.x + threadIdx.x;   // E*64 total (E*64 < 2^31)
  int e = idx >> 6;
  if (e >= E) return;
  int c = (idx & 63) << 2;
  int s = srcv[e], d = dstv[e];
  float nrm = dinv[s] * dinv[d];
  const float4 v = *(const float4*)(hw + (size_t)s * HDIM + c);
  float* ap = agg + (size_t)d * HDIM + c;
  atomicAdd(ap + 0, v.x * nrm);
  atomicAdd(ap + 1, v.y * nrm);
  atomicAdd(ap + 2, v.z * nrm);
  atomicAdd(ap + 3, v.w * nrm);
}

// ---------------- fused BN + ReLU + residual + layer-weighted accumulation -----------
__global__ __launch_bounds__(256)
void post_kernel(const float* __restrict__ agg, const float* __restrict__ hw,
                 const float* __restrict__ selfnorm,
                 const float* __restrict__ conv_b,
                 const float* __restrict__ bn_g, const float* __restrict__ bn_b,
                 const float* __restrict__ bn_rm, const float* __restrict__ bn_rv,
                 const float* __restrict__ x_input,
                 float* __restrict__ h,            // in: layer input (== prev), out: new h
                 float* __restrict__ s_acc,
                 const float* __restrict__ wvec, int layer, int total) {
  int i = blockIdx.x * blockDim.x + threadIdx.x;
  if (i >= total) return;
  int n = i >> 8;                                   // HDIM == 256
  int c = i & (HDIM - 1);
  float pre = agg[i] + hw[i] * selfnorm[n] + conv_b[c];
  float bn  = bn_g[c] * (pre - bn_rm[c]) * rsqrtf(bn_rv[c] + BN_EPS) + bn_b[c];
  float r   = fmaxf(bn, 0.0f);
  float prev = h[i];                                // previous layer output == layer input
  float out  = r + 0.2f * x_input[i] + (layer > 0 ? 0.5f * prev : 0.0f);
  h[i] = out;
  float wl = wvec[layer];
  s_acc[i] = (layer > 0 ? s_acc[i] : 0.0f) + wl * out;
}

// ---------------- output GEMM (K=256 -> C=40) + row-wise log_softmax ------------------
__global__ __launch_bounds__(64)
void out_kernel(const float* __restrict__ s_acc, const float* __restrict__ out_W,
                const float* __restrict__ out_b, float* __restrict__ out, int N) {
  __shared__ float srow[HDIM];
  __shared__ float logits[NCLS];
  __shared__ float lse;
  int n = blockIdx.x;
  int t = threadIdx.x;
#pragma unroll
  for (int j = 0; j < HDIM / 64; ++j)
    srow[t + j * 64] = s_acc[(size_t)n * HDIM + t + j * 64];
  __syncthreads();
  if (t < NCLS) {
    float acc = out_b[t];
#pragma unroll 8
    for (int k = 0; k < HDIM; ++k) acc += srow[k] * out_W[k * NCLS + t];
    logits[t] = acc;
  }
  __syncthreads();
  if (t == 0) {
    float m = -1e30f;
    for (int c2 = 0; c2 < NCLS; ++c2) m = fmaxf(m, logits[c2]);
    float ssum = 0.0f;
    for (int c2 = 0; c2 < NCLS; ++c2) ssum += __expf(logits[c2] - m);
    lse = m + __logf(ssum);
  }
  __syncthreads();
  if (t < NCLS) out[(size_t)n * NCLS + t] = logits[t] - lse;
}

// ======================================================================================
extern "C" void kernel_launch(void* const* d_in, const int* in_sizes, int n_in,
                              void* d_out, int out_size, void* d_ws, size_t ws_size,
                              hipStream_t stream) {
  const float* x      = (const float*)d_in[0];
  const int*   edge   = (const int*)  d_in[1];
  const float* in_W   = (const float*)d_in[2];
  const float* in_b   = (const float*)d_in[3];
  const float* conv_W = (const float*)d_in[4];
  const float* conv_b = (const float*)d_in[5];
  const float* bn_g   = (const float*)d_in[6];
  const float* bn_b   = (const float*)d_in[7];
  const float* bn_rm  = (const float*)d_in[8];
  const float* bn_rv  = (const float*)d_in[9];
  const float* lw     = (const float*)d_in[10];
  const float* out_W  = (const float*)d_in[11];
  const float* out_b  = (const float*)d_in[12];

  const int N = in_sizes[0] / F_IN;     // 50000
  const int E = in_sizes[1] / 2;        // 800000
  const int* srcv = edge;
  const int* dstv = edge + E;

  // workspace carve-out (256B aligned)
  char*  ws  = (char*)d_ws;
  size_t off = 0;
  auto carve = [&](size_t bytes) -> float* {
    float* p = (float*)(ws + off);
    off += (bytes + 255) & ~(size_t)255;
    return p;
  };
  const size_t nhB = (size_t)N * HDIM * sizeof(float);
  float* deg      = carve((size_t)N * sizeof(float));
  float* dinv     = carve((size_t)N * sizeof(float));
  float* selfnorm = carve((size_t)N * sizeof(float));
  float* wvec     = carve(NLAY * sizeof(float));
  float* h        = carve(nhB);
  float* hw       = carve(nhB);
  float* agg      = carve(nhB);
  float* xin      = carve(nhB);
  float* sacc     = carve(nhB);
  (void)ws_size; (void)n_in; (void)out_size;

  // ---- degrees + norms + layer softmax ----
  deg_init_kernel    <<<(N + 255) / 256, 256, 0, stream>>>(deg, N);
  deg_scatter_kernel <<<(E + 255) / 256, 256, 0, stream>>>(dstv, deg, E);
  deg_finalize_kernel<<<(N + 255) / 256, 256, 0, stream>>>(deg, dinv, selfnorm, N);
  layerw_kernel      <<<1, 32, 0, stream>>>(lw, wvec);

  // ---- input transform: h = x @ in_W + in_b ----
  gemm_f32_wmma_kernel<<<(N + 15) / 16, 256, 0, stream>>>(x, in_W, in_b, h, N, F_IN);
  hipMemcpyAsync(xin, h, nhB, hipMemcpyDeviceToDevice, stream);   // x_input = h

  // ---- 6 GCN layers ----
  const int total = N * HDIM;
  for (int l = 0; l < NLAY; ++l) {
    gemm_f32_wmma_kernel<<<(N + 15) / 16, 256, 0, stream>>>(
        h, conv_W + (size_t)l * HDIM * HDIM, nullptr, hw, N, HDIM);
    zero_kernel<<<(total + 255) / 256, 256, 0, stream>>>(agg, total);
    int nthr = E * 64;                                            // edge x 4-channel groups
    edge_scatter_kernel<<<(nthr + 255) / 256, 256, 0, stream>>>(srcv, dstv, dinv, hw, agg, E);
    post_kernel<<<(total + 255) / 256, 256, 0, stream>>>(
        agg, hw, selfnorm,
        conv_b + (size_t)l * HDIM,
        bn_g + (size_t)l * HDIM, bn_b + (size_t)l * HDIM,
        bn_rm + (size_t)l * HDIM, bn_rv + (size_t)l * HDIM,
        xin, h, sacc, wvec, l, total);
  }

  // ---- output head: logits + log_softmax ----
  out_kernel<<<N, 64, 0, stream>>>(sacc, out_W, out_b, (float*)d_out, N);
}